// IRFReconv1D_49692771614765
// MI455X (gfx1250) — compile-verified
//
#include <hip/hip_runtime.h>

// CDNA5 (gfx1250) wave32 WMMA types
typedef __attribute__((ext_vector_type(2))) float v2f;   // 16x4 / 4x16 f32 operand: 2 VGPRs
typedef __attribute__((ext_vector_type(4))) float v4f;
typedef __attribute__((ext_vector_type(8))) float v8f;   // 16x16 f32 C/D: 8 VGPRs

namespace {
constexpr int kW      = 2048;  // signal length per batch row
constexpr int kT      = 64;    // IRF taps
constexpr int kPadBeg = 31;    // (T-1)//2
constexpr int kWaves  = 4;     // independent waves per block, one batch row each
constexpr int kXPad   = 2176;  // 32 left halo + 2048 + right halo (>= index 2112)
constexpr int kChunks = 20;    // contraction s in [0,79] -> 20 WMMA K=4 steps
}

// out[b, w] = sum_{t=0..63} dk[b, w + t - 31] * (irf[b,t] / max_t irf[b,t])
//
// Toeplitz-WMMA form per 256-wide tile:
//   D[i, m] = sum_s A[i, s] * B[s, m]
//   A[i, s] = x[wb + 16*i - 31 + s]           (16 overlapping windows)
//   B[s, m] = k[s - m] if 0 <= s-m < 64 else 0 (banded Toeplitz, per batch)
__global__ __launch_bounds__(kWaves * 32) void irf_reconv1d_wmma(
    const float* __restrict__ dk, const float* __restrict__ irf,
    float* __restrict__ out, int nbatch)
{
  __shared__ __align__(16) float xs[kWaves][kXPad];
  __shared__ float ks[kWaves][kT];

  const int lane  = threadIdx.x & 31;
  const int wave  = threadIdx.x >> 5;
  const int batch = blockIdx.x * kWaves + wave;
  if (batch >= nbatch) return;   // wave-uniform: EXEC stays all-ones for WMMA

  float* __restrict__ x  = xs[wave];
  float* __restrict__ kq = ks[wave];

  // ---- zero the halos (left [0,32), right [2080,2176)) -----------------
  x[lane]        = 0.0f;
  x[2080 + lane] = 0.0f;
  x[2112 + lane] = 0.0f;
  x[2144 + lane] = 0.0f;

  // ---- stage the 2048-sample row into LDS with 16B vector loads --------
  const float* __restrict__ row = dk + (size_t)batch * kW;
#pragma unroll
  for (int j = 0; j < kW / (32 * 4); ++j) {
    const int idx = (j * 32 + lane) * 4;
    *(v4f*)(x + 32 + idx) = *(const v4f*)(row + idx);
  }

  // ---- normalize IRF: k = irf / max(irf) (wave32 shuffle reduction) ----
  const float* __restrict__ kp = irf + (size_t)batch * kT;
  const float k0 = kp[2 * lane];
  const float k1 = kp[2 * lane + 1];
  float mx = fmaxf(k0, k1);
#pragma unroll
  for (int off = 16; off >= 1; off >>= 1)
    mx = fmaxf(mx, __shfl_xor(mx, off, 32));
  kq[2 * lane]     = k0 / mx;
  kq[2 * lane + 1] = k1 / mx;

  // ---- lane roles per ISA 7.12.2 operand layouts -----------------------
  const int half = lane >> 4;   // 0: lanes 0-15, 1: lanes 16-31
  const int n    = lane & 15;   // B/D column N; also A row M
  const int kA   = half * 2;    // .x holds K=kA, .y holds K=kA+1

  // ---- precompute 20 banded-Toeplitz B fragments into registers --------
  v2f bfrag[kChunks];
#pragma unroll
  for (int c = 0; c < kChunks; ++c) {
    const int s0 = 4 * c + kA - n;                       // tap for .x
    const float b0 = (s0 >= 0     && s0 < kT)     ? kq[s0]     : 0.0f;
    const float b1 = (s0 + 1 >= 0 && s0 + 1 < kT) ? kq[s0 + 1] : 0.0f;
    v2f b; b.x = b0; b.y = b1;
    bfrag[c] = b;
  }

  // A element address: x_padded[32 + wb + 16*M + s - kPadBeg], s = 4c + K
  const float* __restrict__ xbase = x + (32 - kPadBeg) + 16 * n + kA;

  // ---- 8 output tiles of 256 samples, 20 accumulating WMMAs each -------
  for (int tile = 0; tile < kW / 256; ++tile) {
    const int wb = tile * 256;
    v8f acc = {};
#pragma unroll
    for (int c = 0; c < kChunks; ++c) {
      v2f a;
      a.x = xbase[wb + 4 * c];
      a.y = xbase[wb + 4 * c + 1];
      // 8 args: (neg_a, A, neg_b, B, c_mod, C, reuse_a, reuse_b)
      acc = __builtin_amdgcn_wmma_f32_16x16x4_f32(
          false, a, false, bfrag[c], (short)0, acc, false, false);
    }
    // D layout: VGPR r holds M = r + 8*half, N = lane&15
    float* __restrict__ op = out + (size_t)batch * kW + wb + half * 128 + n;
#pragma unroll
    for (int r = 0; r < 8; ++r)
      op[r * 16] = acc[r];
  }
}

extern "C" void kernel_launch(void* const* d_in, const int* in_sizes, int n_in,
                              void* d_out, int out_size, void* d_ws, size_t ws_size,
                              hipStream_t stream) {
  const float* dk  = (const float*)d_in[0];  // (B, 2048, 1) f32
  const float* irf = (const float*)d_in[1];  // (B, 64) f32
  float* out = (float*)d_out;                // (B, 2048, 1) f32

  const int nbatch = in_sizes[1] / kT;       // 16384
  dim3 grid((nbatch + kWaves - 1) / kWaves);
  dim3 block(kWaves * 32);
  irf_reconv1d_wmma<<<grid, block, 0, stream>>>(dk, irf, out, nbatch);
}